// TaxonGNNEncoder_5153960755631
// MI455X (gfx1250) — compile-verified
//
#include <hip/hip_runtime.h>

// TaxonGNNEncoder: 2-layer GraphSAGE (mean aggregation) + linear head.
// N=100000 nodes, E=1600000 edges, all feature dims = 256, fp32.
//
// MI455X reasoning: the edge scatter (~6.4 GB of gathers + f32 atomics) is the
// HBM-bound hot spot; the five 100Kx256x256 GEMMs (6.6e10 FLOPs) are cheap by
// comparison, so we keep them in full f32 using V_WMMA_F32_16X16X4_F32 rather
// than downcasting. The two per-layer GEMMs are fused into one K=512 GEMM:
//   mean @ W_l + x @ W_r == [mean | x] @ [W_l ; W_r]
//
// LDS layouts are chosen so every WMMA fragment is a single aligned 8-byte
// conflict-free LDS read, and all global<->LDS staging is b128.

#define D 256

typedef __attribute__((ext_vector_type(2))) float v2f;
typedef __attribute__((ext_vector_type(8))) float v8f;

__global__ void zero_kernel(float* __restrict__ p, long long n) {
  long long i = (long long)blockIdx.x * blockDim.x + threadIdx.x;
  long long stride = (long long)gridDim.x * blockDim.x;
  for (; i < n; i += stride) p[i] = 0.0f;
}

__global__ void deg_kernel(const int* __restrict__ dst, float* __restrict__ deg, int E) {
  int e = blockIdx.x * blockDim.x + threadIdx.x;
  if (e < E) atomicAdd(&deg[dst[e]], 1.0f);
}

__global__ void invdeg_kernel(const float* __restrict__ deg, float* __restrict__ invd, int n) {
  int i = blockIdx.x * blockDim.x + threadIdx.x;
  if (i < n) invd[i] = 1.0f / fmaxf(deg[i], 1.0f);
}

// 64 threads per edge; each thread moves one float4 (coalesced 1KB source row
// reads) and scatters with 4 f32 atomics (global_atomic_add_f32 in L2).
__global__ void scatter_kernel(const int* __restrict__ src, const int* __restrict__ dst,
                               const float* __restrict__ X, float* __restrict__ agg, int E) {
  int tid = blockIdx.x * blockDim.x + threadIdx.x;
  if (tid >= E * 64) return;
  int e  = tid >> 6;
  int c4 = (tid & 63) << 2;
  int s = src[e];
  int d = dst[e];
  const float4 v = *(const float4*)(X + (size_t)s * D + c4);
  float* p = agg + (size_t)d * D + c4;
  atomicAdd(p + 0, v.x);
  atomicAdd(p + 1, v.y);
  atomicAdd(p + 2, v.z);
  atomicAdd(p + 3, v.w);
}

// Fused SAGE layer GEMM (DUAL=1):  Out = act( [agg*invdeg | A1] @ [W0;W1] + bias )
// Plain GEMM head     (DUAL=0):    Out = act( A0 @ W0 + bias )
//
// Block = 256 threads (8 wave32s). Output tile 128x64; each wave owns a 16x64
// strip (4 v8f accumulators). K staged via LDS in chunks of 32.
//
// WMMA f32 16x16x4 fragment layout (ISA 7.12.2):
//   A: lane L -> row M = L%16, K pair {k + 2*(L/16), k+1 + 2*(L/16)}
//   B: lane L -> col N = L%16, same K pair
//   C: VGPR r -> row r + 8*(L/16), col L%16
// ldsA: row-major, stride 36 (A fragment = contiguous K pair -> b64 load;
//       lo/hi half-waves land on disjoint bank sets mod 64).
// ldsB: K-pair interleaved, B[k][n] at ldsB[k/2][2n + (k&1)], row stride 160
//       (== 32 mod 64) -> each B fragment is one aligned b64 load; lo-half
//       covers banks 0..31, hi-half 32..63 -> conflict-free full-bank reads.
template <int DUAL>
__global__ __launch_bounds__(256) void sage_gemm_kernel(
    const float* __restrict__ A0, const float* __restrict__ invd,
    const float* __restrict__ A1, const float* __restrict__ W0,
    const float* __restrict__ W1, const float* __restrict__ bias,
    float* __restrict__ Out, int Nrows, int relu) {
  constexpr int BM = 128, BN = 64, BK = 32;
  constexpr int KT = DUAL ? 2 * D : D;
  constexpr int BSTR = 2 * BN + 32;    // 160 floats, == 32 mod 64

  __shared__ float ldsA[BM][BK + 4];   // 128 x 36 f32
  __shared__ float ldsB[BK / 2][BSTR]; // 16 x 160 f32 (K-pair interleaved)

  const int t    = threadIdx.x;
  const int wave = t >> 5;
  const int lane = t & 31;
  const int m    = lane & 15;      // M (for A) / N (for B) index within 16
  const int hi   = lane >> 4;      // half-wave select
  const int koff = hi * 2;         // fragment K-pair base: lanes16-31 hold K+2,K+3
  const int rowL = wave * 16;
  const int rowBase = blockIdx.x * BM;
  const int colBase = blockIdx.y * BN;

  v8f acc[4] = {};

  for (int k0 = 0; k0 < KT; k0 += BK) {
    __syncthreads();
    // ---- stage A tile 128x32 as float4s (8 float4 per row; 16B-aligned LDS
    //      stores since the 36-float row stride is a multiple of 4 floats).
    //      The DUAL source select is uniform per chunk and hoists.
#pragma unroll
    for (int i = 0; i < (BM * BK) / (256 * 4); ++i) {   // 4 float4 per thread
      int idx = i * 256 + t;
      int r = idx >> 3;            // tile row 0..127
      int c = (idx & 7) << 2;      // col in floats
      int grow = rowBase + r;
      if (grow >= Nrows) grow = Nrows - 1;   // clamp; stores are guarded
      float4 v;
      if (DUAL) {
        if (k0 < D) {              // mean part: scale gathered sum by 1/deg
          v = *(const float4*)(A0 + (size_t)grow * D + (k0 + c));
          float s = invd[grow];
          v.x *= s; v.y *= s; v.z *= s; v.w *= s;
        } else {                   // root part
          v = *(const float4*)(A1 + (size_t)grow * D + (k0 - D + c));
        }
      } else {
        v = *(const float4*)(A0 + (size_t)grow * D + (k0 + c));
      }
      *(float4*)&ldsA[r][c] = v;
    }
    // ---- stage B tile 32x64: each thread loads a K-row pair (2x b128),
    //      interleaves in registers, writes 8 contiguous floats (2x b128)
    //      in the K-pair interleaved layout.
    {
      int kpair = t >> 4;          // 0..15  (K rows 2*kpair, 2*kpair+1)
      int n4    = (t & 15) << 2;   // 0..60
      int gk = k0 + kpair * 2;
      int gn = colBase + n4;
      const float* Wsrc = W0;
      if (DUAL && gk >= D) { Wsrc = W1; gk -= D; }
      float4 r0 = *(const float4*)(Wsrc + (size_t)gk * D + gn);
      float4 r1 = *(const float4*)(Wsrc + (size_t)(gk + 1) * D + gn);
      float* dp = &ldsB[kpair][2 * n4];
      *(float4*)(dp + 0) = make_float4(r0.x, r1.x, r0.y, r1.y);
      *(float4*)(dp + 4) = make_float4(r0.z, r1.z, r0.w, r1.w);
    }
    __syncthreads();

    // ---- 8 x (K=4) WMMA steps per chunk, 4 column tiles per wave
#pragma unroll
    for (int k4 = 0; k4 < BK / 4; ++k4) {
      v2f a = *(const v2f*)&ldsA[rowL + m][k4 * 4 + koff];
      const int kp = k4 * 2 + hi;            // packed-B row for this half-wave
#pragma unroll
      for (int ct = 0; ct < 4; ++ct) {
        v2f b = *(const v2f*)&ldsB[kp][2 * (ct * 16 + m)];
        // D = A(16x4,f32) * B(4x16,f32) + C(16x16,f32)
        acc[ct] = __builtin_amdgcn_wmma_f32_16x16x4_f32(
            false, a, false, b, (short)0, acc[ct], false, false);
      }
    }
  }

  // ---- epilogue: C layout = VGPR r -> row r + 8*hi, col = lane%16
  const int rHalf = hi * 8;
#pragma unroll
  for (int ct = 0; ct < 4; ++ct) {
    int gcol = colBase + ct * 16 + m;
    float bv = bias[gcol];
#pragma unroll
    for (int r = 0; r < 8; ++r) {
      int grow = rowBase + rowL + rHalf + r;
      if (grow < Nrows) {
        float v = acc[ct][r] + bv;
        if (relu) v = fmaxf(v, 0.0f);
        Out[(size_t)grow * D + gcol] = v;
      }
    }
  }
}

extern "C" void kernel_launch(void* const* d_in, const int* in_sizes, int n_in,
                              void* d_out, int out_size, void* d_ws, size_t ws_size,
                              hipStream_t stream) {
  const float* x   = (const float*)d_in[0];
  const int*   ei  = (const int*)d_in[1];   // [2, E] int32
  const float* W1l = (const float*)d_in[2];
  const float* b1  = (const float*)d_in[3];
  const float* W1r = (const float*)d_in[4];
  const float* W2l = (const float*)d_in[5];
  const float* b2  = (const float*)d_in[6];
  const float* W2r = (const float*)d_in[7];
  const float* W3  = (const float*)d_in[8];
  const float* b3  = (const float*)d_in[9];
  float* out = (float*)d_out;

  const int Nn = in_sizes[0] / D;       // 100000
  const int E  = in_sizes[1] / 2;       // 1600000
  const int* src = ei;
  const int* dst = ei + E;

  // workspace layout (f32): agg[N*256] h1[N*256] h2[N*256] deg[N] invd[N]
  float* agg  = (float*)d_ws;
  float* h1   = agg + (size_t)Nn * D;
  float* h2   = h1  + (size_t)Nn * D;
  float* deg  = h2  + (size_t)Nn * D;
  float* invd = deg + Nn;

  const long long nfeat = (long long)Nn * D;
  dim3 blk(256);
  const int zgrid = 2048;
  dim3 ggrid((Nn + 127) / 128, D / 64);

  // degrees are layer-invariant: compute once
  zero_kernel<<<zgrid, blk, 0, stream>>>(deg, Nn);
  deg_kernel<<<(E + 255) / 256, blk, 0, stream>>>(dst, deg, E);
  invdeg_kernel<<<(Nn + 255) / 256, blk, 0, stream>>>(deg, invd, Nn);

  // layer 1
  zero_kernel<<<zgrid, blk, 0, stream>>>(agg, nfeat);
  scatter_kernel<<<(E * 64 + 255) / 256, blk, 0, stream>>>(src, dst, x, agg, E);
  sage_gemm_kernel<1><<<ggrid, blk, 0, stream>>>(agg, invd, x, W1l, W1r, b1, h1, Nn, 1);

  // layer 2
  zero_kernel<<<zgrid, blk, 0, stream>>>(agg, nfeat);
  scatter_kernel<<<(E * 64 + 255) / 256, blk, 0, stream>>>(src, dst, h1, agg, E);
  sage_gemm_kernel<1><<<ggrid, blk, 0, stream>>>(agg, invd, h1, W2l, W2r, b2, h2, Nn, 1);

  // linear head (no relu)
  sage_gemm_kernel<0><<<ggrid, blk, 0, stream>>>(h2, nullptr, nullptr, W3, nullptr, b3, out, Nn, 0);
}